// Rewavelet_24541443129858
// MI455X (gfx1250) — compile-verified
//
#include <hip/hip_runtime.h>
#include <hip/hip_bf16.h>

// Rewavelet: inverse 2x2 Haar synthesis.
// in : (8,256,256,64) f32, channel c = gi*4 + s, subbands s=a,b,c,d
// out: (8,512,512,16) f32
// out[b, 2h+ph, 2w+pw, gi] = a + (-1)^pw*b + (-1)^ph*c + (-1)^(ph+pw)*d
//
// Pure bandwidth-bound (268 MB traffic, ~100 MFLOP): optimize the data path,
// not the math. Strategy: stage through LDS with CDNA5 async global<->LDS
// B128 transfers so that the 2x2 spatial scatter becomes two fully
// contiguous 2 KB output-row stores per block.

// Pointer type the async-LDS builtins expect (from hipcc diagnostic):
// 'int __attribute__((__vector_size__(4 * sizeof(int)))) *' (generic AS)
typedef int vint4_t __attribute__((__vector_size__(16)));

#if defined(__AMDGCN__)
#  if __has_builtin(__builtin_amdgcn_global_load_async_to_lds_b128) && \
      __has_builtin(__builtin_amdgcn_global_store_async_from_lds_b128) && \
      __has_builtin(__builtin_amdgcn_s_wait_asynccnt)
#    define USE_ASYNC_LDS 1
#  else
#    define USE_ASYNC_LDS 0
#    warning "gfx1250 async-LDS builtins not available; using synchronous copy fallback"
#  endif
#else
#  define USE_ASYNC_LDS 0
#endif

namespace {
constexpr int B = 8;
constexpr int H = 256;
constexpr int W = 256;
constexpr int C = 64;        // 16 groups * 4 subbands
constexpr int G = 16;        // groups (output channels)
constexpr int PIX_PER_BLK = 16;  // input pixels per block (one w-chunk)
}  // namespace

__global__ __launch_bounds__(256)
void rewavelet_kernel(const float* __restrict__ in, float* __restrict__ out) {
    // LDS staging: 4 KB input chunk + 4 KB output image (two half-rows)
    __shared__ alignas(16) float ldsIn[PIX_PER_BLK * C];          // [wloc][c]   = 1024 f32
    __shared__ alignas(16) float ldsOut[2 * 2 * PIX_PER_BLK * G]; // [ph][x][gi] = 1024 f32

    const int t   = threadIdx.x;                 // 0..255
    const unsigned blk = blockIdx.x;             // B*H*(W/16) = 32768 blocks
    const int w0  = (int)(blk & 15u) * PIX_PER_BLK;
    const int h   = (int)(blk >> 4) & (H - 1);
    const int b   = (int)(blk >> 12);

    const float* gin = in + ((size_t)((b * H + h) * W + w0) * C);

    // ---- Phase 1: global -> LDS, 4 KB, one async b128 per thread ----
#if USE_ASYNC_LDS
    __builtin_amdgcn_global_load_async_to_lds_b128(
        (vint4_t*)(gin + t * 4),
        (vint4_t*)(&ldsIn[t * 4]),
        /*offset=*/0, /*cpol=*/0);
    __builtin_amdgcn_s_wait_asynccnt(0);
#else
    ((float4*)ldsIn)[t] = ((const float4*)gin)[t];
#endif
    __syncthreads();

    // ---- Phase 2: 4-point butterfly per (pixel, group) ----
    {
        const int wloc = t >> 4;    // 0..15  local pixel
        const int gi   = t & 15;    // 0..15  group
        const float4 v = *(const float4*)(&ldsIn[wloc * C + gi * 4]);
        const float ap = v.x + v.y;   // a+b
        const float am = v.x - v.y;   // a-b
        const float cp = v.z + v.w;   // c+d
        const float cm = v.z - v.w;   // c-d
        const int x0 = 2 * wloc;
        // ldsOut[ph][x][gi], row stride 2*PIX_PER_BLK*G = 512 floats
        ldsOut[0 * 512 + (x0 + 0) * G + gi] = ap + cp;  // ph=0,pw=0
        ldsOut[0 * 512 + (x0 + 1) * G + gi] = am + cm;  // ph=0,pw=1
        ldsOut[1 * 512 + (x0 + 0) * G + gi] = ap - cp;  // ph=1,pw=0
        ldsOut[1 * 512 + (x0 + 1) * G + gi] = am - cm;  // ph=1,pw=1
    }
    __syncthreads();

    // ---- Phase 3: LDS -> global, two contiguous 2 KB rows, async b128 ----
    {
        const int ph = t >> 7;      // 0..1  output row within block
        const int j  = t & 127;     // 0..127 float4 index within row
        const size_t y = (size_t)(2 * h + ph);
        float* gout = out + (((size_t)b * (2 * H) + y) * (2 * W) + 2 * w0) * G;
#if USE_ASYNC_LDS
        __builtin_amdgcn_global_store_async_from_lds_b128(
            (vint4_t*)(gout + j * 4),
            (vint4_t*)(&ldsOut[ph * 512 + j * 4]),
            /*offset=*/0, /*cpol=*/0);
        __builtin_amdgcn_s_wait_asynccnt(0);
#else
        ((float4*)gout)[j] = ((const float4*)ldsOut)[ph * 128 + j];
#endif
    }
}

extern "C" void kernel_launch(void* const* d_in, const int* in_sizes, int n_in,
                              void* d_out, int out_size, void* d_ws, size_t ws_size,
                              hipStream_t stream) {
    (void)in_sizes; (void)n_in; (void)out_size; (void)d_ws; (void)ws_size;
    const float* in = (const float*)d_in[0];
    float* out = (float*)d_out;
    const int blocks = B * H * (W / PIX_PER_BLK);   // 8*256*16 = 32768
    rewavelet_kernel<<<blocks, 256, 0, stream>>>(in, out);
}